// CPCCriterion_43834436223243
// MI455X (gfx1250) — compile-verified
//
#include <hip/hip_runtime.h>
#include <math.h>

#define B_ 16
#define S_ 128
#define D_ 256
#define K_ 12
#define N_ 128
#define W_ (S_ - K_)   // 116

typedef __attribute__((ext_vector_type(2))) float    v2f;
typedef __attribute__((ext_vector_type(8))) float    v8f;
typedef __attribute__((ext_vector_type(4))) unsigned v4u;
typedef __attribute__((ext_vector_type(8))) unsigned v8u;

__global__ void cpc_zero_ws(float* ws) {
    if (threadIdx.x < 32) ws[threadIdx.x] = 0.0f;
}

// One block per (b, w): 8 waves, each wave computes a 16(k) x 16(n) score tile
// with V_WMMA_F32_16X16X4_F32 accumulating over D=256 in steps of 4.
// The 12KB prediction tile is DMA'd into LDS by the Tensor Data Mover.
__launch_bounds__(256, 2)
__global__ void cpc_main(const float* __restrict__ pred,
                         const float* __restrict__ latent,
                         const int*   __restrict__ batch_idx,
                         const int*   __restrict__ seq_idx,
                         float*       __restrict__ ws) {
    __shared__ float sA[16 * D_];   // pred tile, rows 12..15 zero-padded (16 KB)
    __shared__ float sS[16 * N_];   // neg scores (8 KB)
    __shared__ float sPos[16];

    const int blk = blockIdx.x;
    const int b   = blk / W_;
    const int w   = blk % W_;
    const int tid = threadIdx.x;

    const float* predTile = pred + (size_t)(b * W_ + w) * (K_ * D_);

    // ---- Wave 0: TDM async copy of predictions[b,w,:,:] (12 KB, 1-D tile) into LDS.
    if (tid < 32) {
        unsigned long long ga = (unsigned long long)predTile;
        // generic LDS pointer: low 32 bits are the LDS byte offset (aperture scheme)
        unsigned lds_off = (unsigned)(unsigned long long)(size_t)sA;

        // D# group 0: [1:0]=count=1, [63:32]=lds_addr, [120:64]=global_addr, [127:126]=type=2
        v4u g0;
        g0.x = 1u;
        g0.y = lds_off;
        g0.z = (unsigned)ga;
        g0.w = (unsigned)((ga >> 32) & 0x01FFFFFFull) | 0x80000000u;

        // D# group 1: data_size=4B (code 2); tensor_dim0=tile_dim0=3072 elems;
        // tensor_dim1=tile_dim1=1; tensor_dim0_stride=3072; everything else 0.
        v8u g1;
        g1.s0 = 0x00020000u;            // workgroup_mask=0, data_size=2 (4B)
        g1.s1 = (3072u & 0xFFFFu) << 16;// tensor_dim0[15:0] in bits 63:48
        g1.s2 = 1u << 16;               // tensor_dim1[15:0]=1 in bits 95:80
        g1.s3 = 3072u << 16;            // tile_dim0=3072 in bits 127:112
        g1.s4 = 1u;                     // tile_dim1=1, tile_dim2=0
        g1.s5 = 3072u;                  // tensor_dim0_stride[31:0]
        g1.s6 = 0u;
        g1.s7 = 0u;

        asm volatile("tensor_load_to_lds %0, %1" :: "s"(g0), "s"(g1) : "memory");
    }

    // ---- All threads: zero pad rows 12..15 (bytes [12288,16384) — disjoint from TDM dest)
    {
        float4 z = make_float4(0.0f, 0.0f, 0.0f, 0.0f);
        for (int i = K_ * (D_ / 4) + tid; i < 16 * (D_ / 4); i += 256)
            ((float4*)sA)[i] = z;
    }

    const int wave = tid >> 5;
    const int lane = tid & 31;
    const int half = lane >> 4;      // 0: K-offsets {0,1}, 1: K-offsets {2,3}
    const int l15  = lane & 15;
    const int doff = half << 1;

    // ---- Gathered negative row for this lane's output column (stays in L2: latent is 2 MB)
    const int n    = wave * 16 + l15;
    const int flat = (b * N_ + n) * W_ + w;
    const int seq  = (seq_idx[flat] + w) % S_;
    const int ext  = batch_idx[flat] * S_ + seq;
    const float* nrow = latent + (size_t)ext * D_;
    __builtin_prefetch(nrow, 0, 1);                 // global_prefetch_b8: warm gathered row

    if (tid < 32) __builtin_amdgcn_s_wait_tensorcnt(0);   // TDM done before barrier
    __syncthreads();

    const float* arow = sA + l15 * D_;              // A row m = lane&15 (pred k index)

    // ---- 16x16 f32 accumulator, K-loop over D in steps of 4
    v8f c = {};
    #pragma unroll 8
    for (int d0 = 0; d0 < D_; d0 += 4) {
        v2f a  = *(const v2f*)(arow + d0 + doff);   // A[m][d0+doff .. +1]
        v2f bb = *(const v2f*)(nrow + d0 + doff);   // B[doff..][n] = neg[n][d0+doff ..]
        c = __builtin_amdgcn_wmma_f32_16x16x4_f32(
                /*neg_a=*/false, a, /*neg_b=*/false, bb,
                /*c_mod=*/(short)0, c, /*reuse_a=*/false, /*reuse_b=*/false);
    }

    const float inv_d = 1.0f / (float)D_;
    #pragma unroll
    for (int r = 0; r < 8; ++r) {
        int row = half ? (r + 8) : r;               // C layout: vgpr r -> M=r / M=r+8
        sS[row * N_ + wave * 16 + l15] = c[r] * inv_d;
    }

    // ---- Positive scores: 12 dot products of length 256 (16 lanes per k)
    if (tid < K_ * 16) {
        int k = tid >> 4;
        int t = tid & 15;
        const float* prow = sA + k * D_;
        const float* lrow = latent + ((size_t)b * S_ + (size_t)(k + 1 + w)) * D_;
        float p = 0.0f;
        for (int d = t; d < D_; d += 16) p += prow[d] * lrow[d];
        #pragma unroll
        for (int off = 8; off > 0; off >>= 1) p += __shfl_down(p, off, 16);
        if (t == 0) sPos[k] = p * inv_d;
    }
    __syncthreads();

    // ---- Per-k logsumexp over [pos, 128 negatives], accumulate loss/acc
    for (int k = wave; k < K_; k += 8) {
        float pos = sPos[k];
        float v0 = sS[k * N_ + lane];
        float v1 = sS[k * N_ + lane + 32];
        float v2 = sS[k * N_ + lane + 64];
        float v3 = sS[k * N_ + lane + 96];
        float lm = fmaxf(fmaxf(v0, v1), fmaxf(v2, v3));
        #pragma unroll
        for (int off = 16; off > 0; off >>= 1) lm = fmaxf(lm, __shfl_xor(lm, off, 32));
        float mneg = lm;                       // max over negatives
        float mm   = fmaxf(mneg, pos);
        float s = __expf(v0 - mm) + __expf(v1 - mm) + __expf(v2 - mm) + __expf(v3 - mm);
        #pragma unroll
        for (int off = 16; off > 0; off >>= 1) s += __shfl_xor(s, off, 32);
        if (lane == 0) {
            s += __expf(pos - mm);
            float loss = __logf(s) + mm - pos;             // logsumexp - pos
            float accv = (pos >= mneg) ? 1.0f : 0.0f;      // argmax==0 (ties -> index 0)
            atomicAdd(&ws[k],      loss);
            atomicAdd(&ws[K_ + k], accv);
        }
    }
}

__global__ void cpc_finalize(const float* __restrict__ ws, float* __restrict__ out) {
    int t = threadIdx.x;
    if (t < 2 * K_) out[t] = ws[t] * (1.0f / (float)(B_ * W_));
}

extern "C" void kernel_launch(void* const* d_in, const int* in_sizes, int n_in,
                              void* d_out, int out_size, void* d_ws, size_t ws_size,
                              hipStream_t stream) {
    const float* pred   = (const float*)d_in[0];
    const float* latent = (const float*)d_in[1];
    const int*   bidx   = (const int*)d_in[2];
    const int*   sidx   = (const int*)d_in[3];
    float* ws  = (float*)d_ws;
    float* out = (float*)d_out;

    cpc_zero_ws<<<1, 32, 0, stream>>>(ws);
    cpc_main<<<B_ * W_, 256, 0, stream>>>(pred, latent, bidx, sidx, ws);
    cpc_finalize<<<1, 32, 0, stream>>>(ws, out);
}